// EgoCentricCrossAttention_78288663872281
// MI455X (gfx1250) — compile-verified
//
#include <hip/hip_runtime.h>

#define BATCH 8
#define NTOK  4096
#define DIM   1024

typedef float v2f __attribute__((ext_vector_type(2)));
typedef float v8f __attribute__((ext_vector_type(8)));

// ---------------------------------------------------------------------------
// Kernel 1: per-batch argmax(ego_mask) (first index of 1, or 0 if none) and
// gather the ego token rows into E (16 x DIM, rows 8..15 zero-padded so the
// WMMA M=16 tiles are well defined).
// ---------------------------------------------------------------------------
__global__ __launch_bounds__(256) void ego_gather_kernel(
    const float* __restrict__ tokens,   // (B, N, D)
    const int*   __restrict__ mask,     // (B, N)
    float*       __restrict__ E)        // (16, D)
{
    __shared__ int s_idx;
    const int b = blockIdx.x;           // 0..7
    const int t = threadIdx.x;
    if (t == 0) s_idx = NTOK;
    __syncthreads();

    int local = NTOK;
    const int* mrow = mask + (size_t)b * NTOK;
    for (int i = t; i < NTOK; i += 256)
        if (mrow[i] != 0 && i < local) local = i;
    atomicMin(&s_idx, local);
    __syncthreads();

    int idx = s_idx;
    if (idx >= NTOK) idx = 0;           // argmax of all-zero mask -> index 0

    const float* src = tokens + ((size_t)b * NTOK + idx) * DIM;
    for (int i = t; i < DIM; i += 256) {
        E[b * DIM + i]           = src[i];
        E[(b + BATCH) * DIM + i] = 0.0f; // zero rows 8..15
    }
}

// ---------------------------------------------------------------------------
// Kernel 2: C(16 x DIM) = A(16 x DIM) @ W(DIM x DIM), fp32, via
// V_WMMA_F32_16X16X4_F32. One wave per 16x16 output tile, K swept in steps
// of 4. VGPR layouts per CDNA5 ISA 7.12.2:
//   A: lane l, reg j  -> A[l&15][k0 + 2*(l>>4) + j]
//   B: lane l, reg j  -> W[k0 + 2*(l>>4) + j][n0 + (l&15)]
//   C: lane l, reg i  -> C[i + 8*(l>>4)][n0 + (l&15)]
// ---------------------------------------------------------------------------
__global__ __launch_bounds__(32) void gemm16_wmma_kernel(
    const float* __restrict__ A,        // 16 x DIM row-major
    const float* __restrict__ W,        // DIM x DIM row-major
    float*       __restrict__ C)        // 16 x DIM row-major
{
    const int n0 = blockIdx.x * 16;
    const int l  = threadIdx.x;         // 0..31, EXEC all ones (no divergence)
    const int hi = l >> 4;              // 0 or 1
    const int mn = l & 15;

    const float* arow = A + mn * DIM + 2 * hi;
    const float* wcol = W + 2 * hi * DIM + (n0 + mn);

    v8f acc = {};
    #pragma unroll 4
    for (int k0 = 0; k0 < DIM; k0 += 4) {
        v2f a, b;
        a.x = arow[k0];
        a.y = arow[k0 + 1];
        b.x = wcol[(size_t)k0 * DIM];
        b.y = wcol[(size_t)(k0 + 1) * DIM];
        // 8 args: (neg_a, A, neg_b, B, c_mod, C, reuse_a, reuse_b)
        acc = __builtin_amdgcn_wmma_f32_16x16x4_f32(
            false, a, false, b, (short)0, acc, false, false);
    }

    #pragma unroll
    for (int i = 0; i < 8; ++i)
        C[(i + 8 * hi) * DIM + n0 + mn] = acc[i];
}

// ---------------------------------------------------------------------------
// Kernel 3: broadcast out[b, n, :] = R[b, :]. Pure HBM-write-bound: 134 MB.
// Each block handles 8 consecutive rows of one batch; each thread keeps its
// float4 of R in a register and stores it 8x (contiguous 4 KB bursts).
// ---------------------------------------------------------------------------
__global__ __launch_bounds__(256) void bcast_kernel(
    const float* __restrict__ R,        // (16, D) -- rows 0..7 used
    float4*      __restrict__ out)      // (B, N, D) as float4
{
    const int b = blockIdx.y;
    const int t = threadIdx.x;          // 0..255 -> DIM/4 float4 per row
    const float4 r = ((const float4*)(R + (size_t)b * DIM))[t];
    size_t base = ((size_t)b * NTOK + (size_t)blockIdx.x * 8) * (DIM / 4) + t;
    #pragma unroll
    for (int j = 0; j < 8; ++j)
        out[base + (size_t)j * (DIM / 4)] = r;
}

// ---------------------------------------------------------------------------
// Launch. Inputs: tokens_B(f32), ego_mask(int), Wq, Wk, Wv, Wo (f32).
// Wq/Wk are provably unused (softmax over a size-1 axis is identically 1).
// ---------------------------------------------------------------------------
extern "C" void kernel_launch(void* const* d_in, const int* in_sizes, int n_in,
                              void* d_out, int out_size, void* d_ws, size_t ws_size,
                              hipStream_t stream)
{
    const float* tokens = (const float*)d_in[0];
    const int*   mask   = (const int*)d_in[1];
    const float* Wv     = (const float*)d_in[4];
    const float* Wo     = (const float*)d_in[5];

    float* E = (float*)d_ws;            // 16*DIM floats
    float* V = E + 16 * DIM;            // 16*DIM floats
    float* R = V + 16 * DIM;            // 16*DIM floats

    ego_gather_kernel<<<BATCH, 256, 0, stream>>>(tokens, mask, E);
    gemm16_wmma_kernel<<<DIM / 16, 32, 0, stream>>>(E, Wv, V);
    gemm16_wmma_kernel<<<DIM / 16, 32, 0, stream>>>(V, Wo, R);

    dim3 bgrid(NTOK / 8, BATCH);
    bcast_kernel<<<bgrid, 256, 0, stream>>>(R, (float4*)d_out);
}